// Attention_concat_5179730559198
// MI455X (gfx1250) — compile-verified
//
#include <hip/hip_runtime.h>

// Attention_concat, MI455X (gfx1250, wave32).
// Algebraic collapse of the rank-2 attention: never materialize N x N.
//   y[b,c,m] = x[b,c,m] + A0[b,c] + A1[b,c]*ek[b,m]
// with A0 = g/N*(Wv p + bv E), A1 = g/N*(Wv S + bv N), ek = a_k.x + c_k.
// eq/ek pass uses V_WMMA_F32_16X16X4_F32 (f32 matrix engine, full precision).

#define B_     4
#define C_     256
#define N_     4096     // H*W = 64*64
#define INTER_ 64

typedef __attribute__((ext_vector_type(2))) float v2f;
typedef __attribute__((ext_vector_type(8))) float v8f;

// ---------------- K1: fold Wq/Wk through Wc:  a_q = Wq^T wq, c_q = wq.bq ----
__global__ void k1_prep(const float* __restrict__ Wq, const float* __restrict__ bq,
                        const float* __restrict__ Wk, const float* __restrict__ bk,
                        const float* __restrict__ Wc,
                        float* __restrict__ aq, float* __restrict__ ak,
                        float* __restrict__ cqk) {
  const int c = threadIdx.x;                       // 256 threads
  float sq = 0.f, sk = 0.f;
  for (int o = 0; o < INTER_; ++o) {
    sq += Wq[o * C_ + c] * Wc[o];
    sk += Wk[o * C_ + c] * Wc[INTER_ + o];
  }
  aq[c] = sq;
  ak[c] = sk;
  if (c == 0) {
    float q = 0.f, k = 0.f;
    for (int o = 0; o < INTER_; ++o) {
      q += Wc[o] * bq[o];
      k += Wc[INTER_ + o] * bk[o];
    }
    cqk[0] = q;
    cqk[1] = k;
  }
}

// ---------------- K2a: eq/ek = [a_q; a_k] (2x256) @ x[b] (256x4096) ---------
// One wave per (b, 16-col tile): 64 chained v_wmma_f32_16x16x4_f32 over K=256.
// A layout (ISA 7.12.2, 32-bit A 16x4): lane&15 = M, K = k0 + 2*(lane>>4) + comp.
// B layout mirrored: lane&15 = N, same K mapping. D: VGPR i = row M=i (lanes 0-15).
// Only D rows 0,1 are consumed, so A rows 2..15 may carry garbage (ak values):
// unconditional loads, no EXEC divergence in the hot loop.
__global__ void k2a_eqek(const float* __restrict__ x, const float* __restrict__ aq,
                         const float* __restrict__ ak, const float* __restrict__ cqk,
                         float* __restrict__ eq, float* __restrict__ ek) {
  const int wave  = blockIdx.x * (blockDim.x >> 5) + (threadIdx.x >> 5);
  const int b     = wave >> 8;           // 256 n-tiles per batch
  const int n0    = (wave & 255) << 4;
  const int lane  = threadIdx.x & 31;
  const int m     = lane & 15;
  const int half  = lane >> 4;
  const float* __restrict__ xb = x + (size_t)b * C_ * N_;

#if defined(__gfx1250__) && __has_builtin(__builtin_amdgcn_wmma_f32_16x16x4_f32)
  const float* arow = (m == 0) ? aq : ak;   // rows 2..15 replicate ak: D rows 2..15 discarded
  const float* __restrict__ xcol = xb + n0 + m;
  v8f acc = {0.f, 0.f, 0.f, 0.f, 0.f, 0.f, 0.f, 0.f};
#pragma unroll 4
  for (int k0 = 0; k0 < C_; k0 += 4) {
    const int k = k0 + 2 * half;             // even -> 8B-aligned A pair
    const float2 av = *(const float2*)(arow + k);   // global_load_b64, broadcast-friendly
    v2f a, bmat;
    a[0]    = av.x;
    a[1]    = av.y;
    bmat[0] = xcol[(size_t)k * N_];
    bmat[1] = xcol[(size_t)(k + 1) * N_];
    acc = __builtin_amdgcn_wmma_f32_16x16x4_f32(
        /*neg_a=*/false, a, /*neg_b=*/false, bmat,
        /*c_mod=*/(short)0, acc, /*reuse_a=*/false, /*reuse_b=*/false);
  }
  if (lane < 16) {                       // D row 0 -> eq, row 1 -> ek
    eq[b * N_ + n0 + lane] = acc[0] + cqk[0];
    ek[b * N_ + n0 + lane] = acc[1] + cqk[1];
  }
#else
  if (lane < 16) {
    float dq = 0.f, dk = 0.f;
    for (int c = 0; c < C_; ++c) {
      const float xv = xb[(size_t)c * N_ + n0 + lane];
      dq += aq[c] * xv;
      dk += ak[c] * xv;
    }
    eq[b * N_ + n0 + lane] = dq + cqk[0];
    ek[b * N_ + n0 + lane] = dk + cqk[1];
  }
#endif
}

// ---------------- K2b: S[b,c] = sum_n x, p[b,c] = sum_n x*eq (float4) -------
__global__ void k2b_sp(const float* __restrict__ x, const float* __restrict__ eq,
                       float* __restrict__ S, float* __restrict__ p) {
  const int bc = blockIdx.x;             // b*C + c, 1024 blocks x 256 threads
  const int b  = bc >> 8;
  const float4* __restrict__ row4 = (const float4*)(x + (size_t)bc * N_);
  const float4* __restrict__ e4   = (const float4*)(eq + (size_t)b * N_);
  float s = 0.f, pp = 0.f;
  for (int i = threadIdx.x; i < N_ / 4; i += blockDim.x) {
    const float4 xv = row4[i];
    const float4 ev = e4[i];
    s  += (xv.x + xv.y) + (xv.z + xv.w);
    pp += xv.x * ev.x + xv.y * ev.y + xv.z * ev.z + xv.w * ev.w;
  }
  __shared__ float sh1[256];
  __shared__ float sh2[256];
  sh1[threadIdx.x] = s;
  sh2[threadIdx.x] = pp;
  __syncthreads();
  for (int st = 128; st > 0; st >>= 1) {
    if (threadIdx.x < st) {
      sh1[threadIdx.x] += sh1[threadIdx.x + st];
      sh2[threadIdx.x] += sh2[threadIdx.x + st];
    }
    __syncthreads();
  }
  if (threadIdx.x == 0) { S[bc] = sh1[0]; p[bc] = sh2[0]; }
}

// ---------------- K2c: E[b] = sum_n eq[b,n] --------------------------------
__global__ void k2c_E(const float* __restrict__ eq, float* __restrict__ E) {
  const int b = blockIdx.x;
  const float4* __restrict__ e4 = (const float4*)(eq + (size_t)b * N_);
  float s = 0.f;
  for (int i = threadIdx.x; i < N_ / 4; i += blockDim.x) {
    const float4 ev = e4[i];
    s += (ev.x + ev.y) + (ev.z + ev.w);
  }
  __shared__ float sh[256];
  sh[threadIdx.x] = s;
  __syncthreads();
  for (int st = 128; st > 0; st >>= 1) {
    if (threadIdx.x < st) sh[threadIdx.x] += sh[threadIdx.x + st];
    __syncthreads();
  }
  if (threadIdx.x == 0) E[b] = sh[0];
}

// ---------------- K3: A0 = g/N*(Wv p + bv E), A1 = g/N*(Wv S + bv N) -------
__global__ void k3_coef(const float* __restrict__ Wv, const float* __restrict__ bv,
                        const float* __restrict__ gamma,
                        const float* __restrict__ S, const float* __restrict__ p,
                        const float* __restrict__ E,
                        float* __restrict__ A0, float* __restrict__ A1) {
  const int b = blockIdx.x;              // 4 blocks x 256 threads
  const int c = threadIdx.x;
  const float* __restrict__ wrow = Wv + (size_t)c * C_;
  const float* __restrict__ pb = p + b * C_;
  const float* __restrict__ Sb = S + b * C_;
  float s1 = 0.f, sv = 0.f;
  for (int j = 0; j < C_; ++j) {
    const float w = wrow[j];
    s1 += w * pb[j];
    sv += w * Sb[j];
  }
  s1 += bv[c] * E[b];
  sv += bv[c] * (float)N_;
  const float g = gamma[0] * (1.0f / (float)N_);
  A0[b * C_ + c] = g * s1;
  A1[b * C_ + c] = g * sv;
}

// ---------------- K4: y = x + A0[b,c] + A1[b,c]*ek[b,m], float4 stream -----
__global__ void k4_out(const float* __restrict__ x, const float* __restrict__ ek,
                       const float* __restrict__ A0, const float* __restrict__ A1,
                       float* __restrict__ y) {
  const size_t t    = (size_t)blockIdx.x * blockDim.x + threadIdx.x;
  const size_t base = t * 4;
  const int    b    = (int)(base / ((size_t)C_ * N_));
  const size_t rem  = base - (size_t)b * C_ * N_;
  const int    c    = (int)(rem / N_);
  const int    m0   = (int)(rem & (N_ - 1));
  const float4 xv = *(const float4*)(x + base);
  const float4 ev = *(const float4*)(ek + (size_t)b * N_ + m0);
  const float  a0 = A0[b * C_ + c];
  const float  a1 = A1[b * C_ + c];
  float4 o;
  o.x = fmaf(a1, ev.x, a0) + xv.x;
  o.y = fmaf(a1, ev.y, a0) + xv.y;
  o.z = fmaf(a1, ev.z, a0) + xv.z;
  o.w = fmaf(a1, ev.w, a0) + xv.w;
  *(float4*)(y + base) = o;
}

extern "C" void kernel_launch(void* const* d_in, const int* in_sizes, int n_in,
                              void* d_out, int out_size, void* d_ws, size_t ws_size,
                              hipStream_t stream) {
  const float* x     = (const float*)d_in[0];
  const float* Wq    = (const float*)d_in[1];
  const float* bq    = (const float*)d_in[2];
  const float* Wk    = (const float*)d_in[3];
  const float* bk    = (const float*)d_in[4];
  const float* Wc    = (const float*)d_in[5];
  const float* Wv    = (const float*)d_in[6];
  const float* bv    = (const float*)d_in[7];
  const float* gamma = (const float*)d_in[8];
  float* out = (float*)d_out;

  // Workspace layout (floats). All results fully written each call; no init needed.
  float* ws  = (float*)d_ws;
  float* aq  = ws;                         // 256 (16B aligned)
  float* ak  = ws + 256;                   // 256 (16B aligned)
  float* cqk = ws + 512;                   // 2
  float* eq  = ws + 1024;                  // B*N = 16384
  float* ek  = ws + 1024 + B_ * N_;        // 16384
  float* S   = ek + B_ * N_;               // B*C = 1024
  float* p   = S + B_ * C_;                // 1024
  float* E   = p + B_ * C_;                // 4
  float* A0  = E + 8;                      // 1024
  float* A1  = A0 + B_ * C_;               // 1024  (~150 KB total)

  k1_prep<<<1, C_, 0, stream>>>(Wq, bq, Wk, bk, Wc, aq, ak, cqk);

  // B * (N/16) = 1024 waves; 8 waves (256 threads) per block -> 128 blocks.
  k2a_eqek<<<(B_ * (N_ / 16)) / 8, 256, 0, stream>>>(x, aq, ak, cqk, eq, ek);

  k2b_sp<<<B_ * C_, 256, 0, stream>>>(x, eq, S, p);
  k2c_E<<<B_, 256, 0, stream>>>(eq, E);
  k3_coef<<<B_, C_, 0, stream>>>(Wv, bv, gamma, S, p, E, A0, A1);

  // B*C*N / 4 float4 lanes / 256 threads = 4096 blocks.
  k4_out<<<(B_ * C_ * N_ / 4) / 256, 256, 0, stream>>>(x, ek, A0, A1, out);
}